// Sampler_5188320493897
// MI455X (gfx1250) — compile-verified
//
#include <hip/hip_runtime.h>
#include <hip/hip_bf16.h>

#define BLOCK 1024
#define NBINS 4096

// Monotone float -> uint32 key: larger float <=> larger key.
__device__ __forceinline__ unsigned int fkey(float f) {
  unsigned int u = __float_as_uint(f);
  return (u & 0x80000000u) ? ~u : (u | 0x80000000u);
}

// Stream one row with 4 independent b128 loads in flight per thread.
template <typename F>
__device__ __forceinline__ void scan_row(const float* __restrict__ L, int V, int tid, F&& f) {
  const int S = BLOCK * 4;          // floats covered per block-wide float4 step
  int i = tid * 4;
  for (; i + 3 * S < V; i += 4 * S) {
    float4 a = *reinterpret_cast<const float4*>(L + i);
    float4 b = *reinterpret_cast<const float4*>(L + i + S);
    float4 c = *reinterpret_cast<const float4*>(L + i + 2 * S);
    float4 d = *reinterpret_cast<const float4*>(L + i + 3 * S);
    f(i, a); f(i + S, b); f(i + 2 * S, c); f(i + 3 * S, d);
  }
  for (; i < V; i += S) {
    float4 a = *reinterpret_cast<const float4*>(L + i);
    f(i, a);
  }
}

__global__ __launch_bounds__(BLOCK) void sampler_kernel(
    const float* __restrict__ logits,
    const float* __restrict__ temps,
    const int*   __restrict__ topks,
    const float* __restrict__ noise,
    int* __restrict__ out, int V)
{
  const int row = blockIdx.x;
  const int tid = (int)threadIdx.x;
  const float* __restrict__ L = logits + (size_t)row * (size_t)V;
  const float* __restrict__ N = noise  + (size_t)row * (size_t)V;

  __shared__ unsigned int hist[NBINS];
  __shared__ unsigned long long gmax;   // packed (logit key, ~index)
  __shared__ unsigned long long smax;   // packed (score key, ~index)
  __shared__ unsigned int s_digit;
  __shared__ int s_krem;
  __shared__ float nbuf[4][BLOCK * 4];  // 4-deep async noise staging ring (64 KB)

  const float temp = temps[row];
  const int   k    = topks[row];
  const bool  greedy  = (temp == 0.0f);
  const bool  do_topk = (!greedy) && (k > 0);

  if (tid == 0) { gmax = 0ull; smax = 0ull; }
  for (int i = tid; i < NBINS; i += BLOCK) hist[i] = 0u;
  __syncthreads();

  // ---- Pass 1 (HBM read, fills L2): greedy argmax + MSD-12 histogram ----
  {
    unsigned long long lmax = 0ull;
    scan_row(L, V, tid, [&](int i, float4 v) {
      const float vv[4] = {v.x, v.y, v.z, v.w};
#pragma unroll
      for (int j = 0; j < 4; ++j) {
        unsigned int key = fkey(vv[j]);
        unsigned long long p = ((unsigned long long)key << 32) |
                               (unsigned long long)(0xFFFFFFFFu - (unsigned)(i + j));
        if (p > lmax) lmax = p;
        if (do_topk) atomicAdd(&hist[key >> 20], 1u);
      }
    });
    atomicMax(&gmax, lmax);
  }
  __syncthreads();

  if (greedy) {  // uniform per-row branch
    if (tid == 0) out[row] = (int)(0xFFFFFFFFu - (unsigned)(gmax & 0xFFFFFFFFull));
    return;
  }

  // ---- Radix-select exact k-th largest key (L2-resident re-reads) ----
  unsigned int T = 0u;  // k==0: keep everything (key >= 0 always true)
  if (do_topk) {
    int krem = (k < 1) ? 1 : ((k > V) ? V : k);
    if (tid == 0) {
      unsigned cum = 0, cb = 0; int b;
      for (b = NBINS - 1; b >= 0; --b) { cb = hist[b]; cum += cb; if (cum >= (unsigned)krem) break; }
      if (b < 0) b = 0;
      s_digit = (unsigned)b; s_krem = krem - (int)(cum - cb);
    }
    __syncthreads();
    unsigned prefix = s_digit << 20; krem = s_krem;

    // bits [19:8]
    for (int i = tid; i < NBINS; i += BLOCK) hist[i] = 0u;
    __syncthreads();
    scan_row(L, V, tid, [&](int i, float4 v) {
      const float vv[4] = {v.x, v.y, v.z, v.w};
#pragma unroll
      for (int j = 0; j < 4; ++j) {
        unsigned key = fkey(vv[j]);
        if ((key >> 20) == (prefix >> 20)) atomicAdd(&hist[(key >> 8) & 0xFFFu], 1u);
      }
    });
    __syncthreads();
    if (tid == 0) {
      unsigned cum = 0, cb = 0; int b;
      for (b = NBINS - 1; b >= 0; --b) { cb = hist[b]; cum += cb; if (cum >= (unsigned)krem) break; }
      if (b < 0) b = 0;
      s_digit = (unsigned)b; s_krem = krem - (int)(cum - cb);
    }
    __syncthreads();
    prefix |= s_digit << 8; krem = s_krem;

    // bits [7:0]
    for (int i = tid; i < 256; i += BLOCK) hist[i] = 0u;
    __syncthreads();
    scan_row(L, V, tid, [&](int i, float4 v) {
      const float vv[4] = {v.x, v.y, v.z, v.w};
#pragma unroll
      for (int j = 0; j < 4; ++j) {
        unsigned key = fkey(vv[j]);
        if ((key >> 8) == (prefix >> 8)) atomicAdd(&hist[key & 0xFFu], 1u);
      }
    });
    __syncthreads();
    if (tid == 0) {
      unsigned cum = 0, cb = 0; int b;
      for (b = 255; b >= 0; --b) { cb = hist[b]; cum += cb; if (cum >= (unsigned)krem) break; }
      if (b < 0) b = 0;
      s_digit = (unsigned)b;
    }
    __syncthreads();
    T = prefix | s_digit;
  }

  // ---- Pass 4: Gumbel-max sampling. Noise streamed HBM->LDS via async DMA,
  //      4-deep ring buffer, prefetch depth 3 (ASYNCcnt-pipelined). ----
  const float invT = 1.0f / temp;
  const int CH = BLOCK * 4;                 // 4096 floats per chunk (16 KB)
  const int nchunk = (V + CH - 1) / CH;

  auto stage = [&](int c, int buf) {
    int idx = c * CH + tid * 4;
    if (idx > V - 4) idx = V - 4;           // clamp keeps all lanes active (uniform ASYNCcnt)
    unsigned lds = (unsigned)(size_t)(void*)&nbuf[buf][tid * 4];
    unsigned long long ga = (unsigned long long)(size_t)(const void*)(N + idx);
    asm volatile("global_load_async_to_lds_b128 %0, %1, off"
                 :: "v"(lds), "v"(ga) : "memory");
  };

  for (int c0 = 0; c0 < 3 && c0 < nchunk; ++c0) stage(c0, c0 & 3);

  unsigned long long lsm = 0ull;
  for (int c = 0; c < nchunk; ++c) {
    // Issue the (L2-resident) logits load before waiting on the async stream.
    int base = c * CH + tid * 4;
    const bool act = base < V;
    float4 lg = {0.f, 0.f, 0.f, 0.f};
    if (act) lg = *reinterpret_cast<const float4*>(L + base);

    if (c + 3 < nchunk) stage(c + 3, (c + 3) & 3);

    int w = nchunk - 1 - c; if (w > 3) w = 3;   // outstanding-after-wait bound; chunk c done
    if      (w == 3) asm volatile("s_wait_asynccnt 0x3" ::: "memory");
    else if (w == 2) asm volatile("s_wait_asynccnt 0x2" ::: "memory");
    else if (w == 1) asm volatile("s_wait_asynccnt 0x1" ::: "memory");
    else             asm volatile("s_wait_asynccnt 0x0" ::: "memory");

    if (act) {
      float4 nz = *reinterpret_cast<const float4*>(&nbuf[c & 3][tid * 4]); // own staged slot
      const float lv[4] = {lg.x, lg.y, lg.z, lg.w};
      const float nv[4] = {nz.x, nz.y, nz.z, nz.w};
#pragma unroll
      for (int j = 0; j < 4; ++j) {
        unsigned key = fkey(lv[j]);
        float score = (key >= T) ? (lv[j] * invT - __logf(nv[j]))
                                 : __int_as_float(0xFF800000u);        // -inf
        unsigned long long p = ((unsigned long long)fkey(score) << 32) |
                               (unsigned long long)(0xFFFFFFFFu - (unsigned)(base + j));
        if (p > lsm) lsm = p;
      }
    }
  }
  atomicMax(&smax, lsm);
  __syncthreads();
  if (tid == 0) out[row] = (int)(0xFFFFFFFFu - (unsigned)(smax & 0xFFFFFFFFull));
}

extern "C" void kernel_launch(void* const* d_in, const int* in_sizes, int n_in,
                              void* d_out, int out_size, void* d_ws, size_t ws_size,
                              hipStream_t stream) {
  const float* logits = (const float*)d_in[0];
  const float* temps  = (const float*)d_in[1];
  const int*   topks  = (const int*)d_in[2];
  const float* noise  = (const float*)d_in[3];
  const int B = in_sizes[1];              // 256 rows
  const int V = in_sizes[0] / B;          // 128000 vocab
  int* out = (int*)d_out;
  (void)n_in; (void)out_size; (void)d_ws; (void)ws_size;
  hipLaunchKernelGGL(sampler_kernel, dim3(B), dim3(BLOCK), 0, stream,
                     logits, temps, topks, noise, out, V);
}